// hashedLayer_38826504356572
// MI455X (gfx1250) — compile-verified
//
#include <hip/hip_runtime.h>
#include <stdint.h>

// ---------------------------------------------------------------------------
// HashedNet layer on MI455X (gfx1250):  out[b,i] = sum_j a[b,j] * W[h(i,j)]
// = fp32 GEMM (M=64, N=8192, K=8192) with B gathered from a 2MB bucket table.
// Engine: V_WMMA_F32_16X16X4_F32, one B fragment reused by 4 M-tile WMMAs.
//
// Roofline: hash_idx stream = 256 MB (~11 us @ 23.3 TB/s); W gathers = 64M
// random 4B loads into an L2-resident 2 MB table -> gather-issue bound, so
// both gathers must be SADDR + scale_offset with zero address VALU.
// ---------------------------------------------------------------------------

typedef __attribute__((ext_vector_type(2))) float v2f;
typedef __attribute__((ext_vector_type(8))) float v8f;
typedef __attribute__((ext_vector_type(2))) int   v2i;

#define FAN_IN    8192
#define FAN_OUT   8192
#define BATCH     64
#define KSPLIT    4                      // K-split waves per N-tile
#define NT_PER_WG 2                      // N-tiles per workgroup
#define KCHUNK    (FAN_IN / KSPLIT)      // 2048
#define SLOT_STRIDE (33 * 32)            // 33-float lane stride -> conflict-free LDS

static __device__ __forceinline__ v8f wmma_f32(v2f a, v2f b, v8f c) {
    // D = A(16x4) * B(4x16) + C(16x16), fp32
    return __builtin_amdgcn_wmma_f32_16x16x4_f32(
        /*neg_a=*/false, a, /*neg_b=*/false, b,
        /*c_mod=*/(short)0, c, /*reuse_a=*/false, /*reuse_b=*/false);
}

// Gather with a 32-bit element index on a uniform base. Keeping the index as
// an independent u32 register and the (idx << 2) shl visible lets the backend
// emit:  global_load_b32 v, v_idx, s[Wbase] scale_offset   (zero VALU).
static __device__ __forceinline__ float gatherW(const float* __restrict__ W,
                                                uint32_t idx) {
    const uint32_t boff = idx << 2;      // idx < 2^21, cannot overflow
    return *(const float*)((const char*)W + boff);
}

__global__ __launch_bounds__(256)
void hashed_layer_wmma(const float* __restrict__ a,
                       const float* __restrict__ W,
                       const int*   __restrict__ hidx,
                       float*       __restrict__ out)
{
    __shared__ float red[(KSPLIT - 1) * NT_PER_WG * SLOT_STRIDE];

    const int tid    = threadIdx.x;
    const int lane   = tid & 31;          // wave32
    const int wave   = tid >> 5;          // 0..7
    const int lane16 = lane & 15;         // N (B,C) / M (A) index within tile
    const int half   = lane >> 4;         // selects K pair {0,1} vs {2,3}

    const int nsub = wave & (NT_PER_WG - 1);   // which N-tile of this WG
    const int ks   = wave >> 1;                // K-split id 0..3
    const int n0   = (blockIdx.x * NT_PER_WG + nsub) * 16;

    const int kbeg = ks * KCHUNK;
    const int kend = kbeg + KCHUNK;

    // Per-lane row bases. hash_idx rows are contiguous in j (the K axis), so
    // each lane's 8B index load is contiguous and its cacheline is reused
    // across consecutive K-steps.
    const v2i* hrow =
        (const v2i*)(hidx + (long long)(n0 + lane16) * FAN_IN) + half;
    const float* arow = a + lane16 * FAN_IN + half * 2;

    v8f c0 = {}, c1 = {}, c2 = {}, c3 = {};

#pragma unroll 8
    for (int k = kbeg; k < kend; k += 4) {
        // ---- B fragment: stream index pair non-temporally (don't thrash the
        //      L2 lines holding the hot 2MB W table); both gathers are
        //      SADDR + scale_offset (no address math).
        const v2i ip = __builtin_nontemporal_load(hrow + (k >> 1));
        v2f b;
        b.x = gatherW(W, (uint32_t)ip.x);
        b.y = gatherW(W, (uint32_t)ip.y);

        // ---- A fragments: one b64 load per M-tile (same addresses across
        //      waves of the same K-split -> WGP$ hits; immediate offsets).
        v2f a0 = *(const v2f*)(arow + 0 * 16 * FAN_IN + k);
        v2f a1 = *(const v2f*)(arow + 1 * 16 * FAN_IN + k);
        v2f a2 = *(const v2f*)(arow + 2 * 16 * FAN_IN + k);
        v2f a3 = *(const v2f*)(arow + 3 * 16 * FAN_IN + k);

        c0 = wmma_f32(a0, b, c0);
        c1 = wmma_f32(a1, b, c1);
        c2 = wmma_f32(a2, b, c2);
        c3 = wmma_f32(a3, b, c3);
    }

    // ------------------------------------------------------------------
    // Deterministic K-split reduction through LDS (lane stride 33 floats:
    // 33*l mod 64 is a permutation -> conflict-free ds traffic).
    // ------------------------------------------------------------------
    if (ks > 0) {
        float* slot = red + ((ks - 1) * NT_PER_WG + nsub) * SLOT_STRIDE + lane * 33;
#pragma unroll
        for (int r = 0; r < 8; ++r) {
            slot[0 * 8 + r] = c0[r];
            slot[1 * 8 + r] = c1[r];
            slot[2 * 8 + r] = c2[r];
            slot[3 * 8 + r] = c3[r];
        }
    }
    __syncthreads();

    if (ks == 0) {
#pragma unroll
        for (int s = 1; s < KSPLIT; ++s) {
            const float* slot =
                red + ((s - 1) * NT_PER_WG + nsub) * SLOT_STRIDE + lane * 33;
#pragma unroll
            for (int r = 0; r < 8; ++r) {
                c0[r] += slot[0 * 8 + r];
                c1[r] += slot[1 * 8 + r];
                c2[r] += slot[2 * 8 + r];
                c3[r] += slot[3 * 8 + r];
            }
        }
        // C layout: vgpr r, lanes 0-15 -> M=r, lanes 16-31 -> M=8+r; N=lane16.
        float* orow = out + (long long)(8 * half) * FAN_OUT + n0 + lane16;
#pragma unroll
        for (int r = 0; r < 8; ++r) {
            __builtin_nontemporal_store(c0[r], orow + (0 * 16 + r) * FAN_OUT);
            __builtin_nontemporal_store(c1[r], orow + (1 * 16 + r) * FAN_OUT);
            __builtin_nontemporal_store(c2[r], orow + (2 * 16 + r) * FAN_OUT);
            __builtin_nontemporal_store(c3[r], orow + (3 * 16 + r) * FAN_OUT);
        }
    }
}

extern "C" void kernel_launch(void* const* d_in, const int* in_sizes, int n_in,
                              void* d_out, int out_size, void* d_ws, size_t ws_size,
                              hipStream_t stream) {
    (void)in_sizes; (void)n_in; (void)d_ws; (void)ws_size; (void)out_size;
    const float* a    = (const float*)d_in[0];   // [64, 8192] fp32
    const float* W    = (const float*)d_in[1];   // [524288]  fp32
    const int*   hidx = (const int*)d_in[2];     // [8192, 8192] int32
    float*       out  = (float*)d_out;           // [64, 8192] fp32

    dim3 grid(FAN_OUT / (16 * NT_PER_WG));       // 256 workgroups
    dim3 block(256);                             // 8 waves: 2 N-tiles x 4 K-splits
    hashed_layer_wmma<<<grid, block, 0, stream>>>(a, W, hidx, out);
}